// TrafficGNN_60438779789903
// MI455X (gfx1250) — compile-verified
//
#include <hip/hip_runtime.h>

typedef float v2f __attribute__((ext_vector_type(2)));
typedef float v8f __attribute__((ext_vector_type(8)));

// ---------------------------------------------------------------------------
// init: deg = 1 (self-loop), agg1 = 0, out = 0.  Re-run every call (harness
// poisons buffers and does not restore between replays).
// ---------------------------------------------------------------------------
__global__ void gcn_init(float* __restrict__ agg1, float* __restrict__ deg,
                         float* __restrict__ out, int N) {
  int i = blockIdx.x * blockDim.x + threadIdx.x;
  if (i < N * 16) agg1[i] = 0.0f;
  if (i < N)      deg[i]  = 1.0f;   // self-loop contribution to in-degree
  if (i < N * 3)  out[i]  = 0.0f;
}

// in-degree over dst (real edges); deg array is L2-resident (400 KB)
__global__ void gcn_degree(const int* __restrict__ dst, float* __restrict__ deg, int E) {
  int e = blockIdx.x * blockDim.x + threadIdx.x;
  if (e < E) atomicAdd(&deg[dst[e]], 1.0f);
}

__global__ void gcn_dinv(const float* __restrict__ deg, float* __restrict__ dinv, int N) {
  int i = blockIdx.x * blockDim.x + threadIdx.x;
  if (i < N) dinv[i] = rsqrtf(deg[i]);   // deg >= 1 always (self-loop)
}

// ---------------------------------------------------------------------------
// GEMM1: h1 = x @ W1.  x:[N,5], W1:[5,16], h1:[N,16].
// One wave per 16-row tile; V_WMMA_F32_16X16X4_F32, K=5 padded to 8 -> 2 WMMAs.
// A layout (16x4 f32): lanes 0-15 hold (K0,K1), lanes 16-31 hold (K2,K3).
// D layout (16x16 f32): VGPR v -> row v (lanes 0-15) / row v+8 (lanes 16-31).
// ---------------------------------------------------------------------------
__global__ void gcn_gemm1(const float* __restrict__ x, const float* __restrict__ W1,
                          float* __restrict__ h1, int N) {
  int wave = (blockIdx.x * blockDim.x + threadIdx.x) >> 5;
  int lane = threadIdx.x & 31;
  int tiles = (N + 15) >> 4;
  if (wave >= tiles) return;               // wave-uniform: EXEC all-1s at WMMA
  int row0 = wave << 4;
  int half = lane >> 4;                    // 0: K pair (0,1) | 1: K pair (2,3)
  int l    = lane & 15;
  int r = row0 + l; if (r >= N) r = N - 1; // clamp (tiles are full for N=100k)
  const float* xr = x + (size_t)r * 5;

  v2f a0, a1, b0, b1;
  // pass 0: K = 0..3 (all valid, K dim = 5)
  a0.x = xr[half ? 2 : 0];
  a0.y = xr[half ? 3 : 1];
  // pass 1: K = 4..7 (only K=4 valid, rest zero-padded)
  a1.x = half ? 0.0f : xr[4];
  a1.y = 0.0f;
  // B (4x16): mirror of A's K striping; W1 row-major [5][16]
  b0.x = W1[(half ? 2 : 0) * 16 + l];
  b0.y = W1[(half ? 3 : 1) * 16 + l];
  b1.x = half ? 0.0f : W1[4 * 16 + l];
  b1.y = 0.0f;

  v8f c = {};
  c = __builtin_amdgcn_wmma_f32_16x16x4_f32(false, a0, false, b0, (short)0, c, false, false);
  c = __builtin_amdgcn_wmma_f32_16x16x4_f32(false, a1, false, b1, (short)0, c, false, false);

  float* hr = h1 + (size_t)row0 * 16 + (size_t)half * 8 * 16 + l;
  if (row0 + 16 <= N) {
    // full tile: unguarded, clause-able coalesced stores (64B per row)
#pragma unroll
    for (int v = 0; v < 8; ++v) hr[v * 16] = c[v];
  } else {
#pragma unroll
    for (int v = 0; v < 8; ++v) {
      int rr = row0 + v + half * 8;
      if (rr < N) h1[(size_t)rr * 16 + l] = c[v];
    }
  }
}

// layer-1 scatter: one thread per (edge, feature); agg1 (6.4 MB) lives in L2
__global__ void gcn_scatter1(const int* __restrict__ src, const int* __restrict__ dst,
                             const float* __restrict__ dinv, const float* __restrict__ h1,
                             float* __restrict__ agg1, int E) {
  long long t = (long long)blockIdx.x * blockDim.x + threadIdx.x;
  int e = (int)(t >> 4);
  if (e >= E) return;
  int f = (int)(t & 15);
  int s = src[e], d = dst[e];
  float w = dinv[s] * dinv[d];
  atomicAdd(&agg1[(size_t)d * 16 + f], h1[(size_t)s * 16 + f] * w);
}

// ---------------------------------------------------------------------------
// GEMM2: h2 = relu(agg1 + dinv^2*h1 + b1) @ W2.   K=16 -> 4 chained WMMAs.
// Layer-1 self-loop, bias and ReLU are fused into the A-operand load.
// W2:[16,3] padded to N=16 columns; only cols 0-2 stored.
// ---------------------------------------------------------------------------
__global__ void gcn_gemm2(const float* __restrict__ agg1, const float* __restrict__ h1,
                          const float* __restrict__ dinv, const float* __restrict__ b1,
                          const float* __restrict__ W2, float* __restrict__ h2, int N) {
  int wave = (blockIdx.x * blockDim.x + threadIdx.x) >> 5;
  int lane = threadIdx.x & 31;
  int tiles = (N + 15) >> 4;
  if (wave >= tiles) return;
  int row0 = wave << 4;
  int half = lane >> 4;
  int l    = lane & 15;
  int r = row0 + l; if (r >= N) r = N - 1;
  float di = dinv[r];
  float sl = di * di;                      // self-loop norm weight
  const float* ag = agg1 + (size_t)r * 16;
  const float* hh = h1   + (size_t)r * 16;

  v8f c = {};
#pragma unroll
  for (int kb = 0; kb < 16; kb += 4) {
    int k0 = kb + (half ? 2 : 0);
    int k1 = k0 + 1;
    float a0v = ag[k0] + sl * hh[k0] + b1[k0];
    float a1v = ag[k1] + sl * hh[k1] + b1[k1];
    v2f a, b;
    a.x = a0v > 0.0f ? a0v : 0.0f;         // fused ReLU
    a.y = a1v > 0.0f ? a1v : 0.0f;
    b.x = (l < 3) ? W2[k0 * 3 + l] : 0.0f; // pad N 3 -> 16
    b.y = (l < 3) ? W2[k1 * 3 + l] : 0.0f;
    c = __builtin_amdgcn_wmma_f32_16x16x4_f32(false, a, false, b, (short)0, c, false, false);
  }
  if (l < 3) {
    if (row0 + 16 <= N) {
      float* hr = h2 + ((size_t)row0 + (size_t)half * 8) * 3 + l;
#pragma unroll
      for (int v = 0; v < 8; ++v) hr[v * 3] = c[v];
    } else {
#pragma unroll
      for (int v = 0; v < 8; ++v) {
        int rr = row0 + v + half * 8;
        if (rr < N) h2[(size_t)rr * 3 + l] = c[v];
      }
    }
  }
}

// layer-2 scatter: (edge, feature) with f in 0..3, f==3 idle; out (1.2 MB) in L2
__global__ void gcn_scatter2(const int* __restrict__ src, const int* __restrict__ dst,
                             const float* __restrict__ dinv, const float* __restrict__ h2,
                             float* __restrict__ out, int E) {
  long long t = (long long)blockIdx.x * blockDim.x + threadIdx.x;
  int e = (int)(t >> 2);
  int f = (int)(t & 3);
  if (e >= E || f >= 3) return;
  int s = src[e], d = dst[e];
  float w = dinv[s] * dinv[d];
  atomicAdd(&out[(size_t)d * 3 + f], h2[(size_t)s * 3 + f] * w);
}

// final: out += b2 + dinv^2 * h2  (layer-2 self-loop + bias), after scatter2
__global__ void gcn_final(const float* __restrict__ h2, const float* __restrict__ dinv,
                          const float* __restrict__ b2, float* __restrict__ out, int N) {
  int t = blockIdx.x * blockDim.x + threadIdx.x;
  if (t >= N * 3) return;
  int i = t / 3;
  int f = t - i * 3;
  float di = dinv[i];
  out[t] += b2[f] + di * di * h2[t];
}

// ---------------------------------------------------------------------------
extern "C" void kernel_launch(void* const* d_in, const int* in_sizes, int n_in,
                              void* d_out, int out_size, void* d_ws, size_t ws_size,
                              hipStream_t stream) {
  const float* x  = (const float*)d_in[0];
  const int*   ei = (const int*)d_in[1];
  const float* W1 = (const float*)d_in[2];
  const float* b1 = (const float*)d_in[3];
  const float* W2 = (const float*)d_in[4];
  const float* b2 = (const float*)d_in[5];
  float* out = (float*)d_out;

  int N = in_sizes[0] / 5;
  int E = in_sizes[1] / 2;
  const int* src = ei;        // edge_index[0]
  const int* dst = ei + E;    // edge_index[1]

  // workspace carve-up (256B aligned blocks); total ~15 MB
  size_t Np = ((size_t)N + 63) & ~(size_t)63;
  float* deg  = (float*)d_ws;            // Np
  float* dinv = deg  + Np;               // Np
  float* h1   = dinv + Np;               // Np*16
  float* agg1 = h1   + Np * 16;          // Np*16
  float* h2   = agg1 + Np * 16;          // Np*4 (uses 3)

  int initT = N * 16;
  gcn_init<<<(initT + 255) / 256, 256, 0, stream>>>(agg1, deg, out, N);
  gcn_degree<<<(E + 255) / 256, 256, 0, stream>>>(dst, deg, E);
  gcn_dinv<<<(N + 255) / 256, 256, 0, stream>>>(deg, dinv, N);

  int tiles = (N + 15) / 16;
  int gT = tiles * 32;                   // one wave32 per 16-row tile
  gcn_gemm1<<<(gT + 255) / 256, 256, 0, stream>>>(x, W1, h1, N);

  long long s1 = (long long)E * 16;
  gcn_scatter1<<<(unsigned)((s1 + 255) / 256), 256, 0, stream>>>(src, dst, dinv, h1, agg1, E);

  gcn_gemm2<<<(gT + 255) / 256, 256, 0, stream>>>(agg1, h1, dinv, b1, W2, h2, N);

  long long s2 = (long long)E * 4;
  gcn_scatter2<<<(unsigned)((s2 + 255) / 256), 256, 0, stream>>>(src, dst, dinv, h2, out, E);

  gcn_final<<<(N * 3 + 255) / 256, 256, 0, stream>>>(h2, dinv, b2, out, N);
}